// GNN_9698036155133
// MI455X (gfx1250) — compile-verified
//
#include <hip/hip_runtime.h>

typedef __attribute__((ext_vector_type(2))) float v2f;
typedef __attribute__((ext_vector_type(8))) float v8f;

#define DIM 128

// ---------------- workspace zero ----------------
__global__ void GNN_zero_kernel(int* __restrict__ p, int n) {
    int i = blockIdx.x * blockDim.x + threadIdx.x;
    if (i < n) p[i] = 0;
}

// ---------------- per-node in-degree ----------------
__global__ void GNN_count_kernel(const long long* __restrict__ col,
                                 int* __restrict__ counts, int E) {
    int e = blockIdx.x * blockDim.x + threadIdx.x;
    if (e < E) atomicAdd(&counts[(int)col[e]], 1);
}

// ---------------- single-workgroup exclusive scan (N ~ 10000) ----------------
__global__ void GNN_scan_kernel(const int* __restrict__ counts,
                                int* __restrict__ offsets, int n) {
    __shared__ int part[1024];
    const int t = threadIdx.x;
    const int CH = (n + 1023) / 1024;
    const int start = t * CH;
    int s = 0;
    for (int k = 0; k < CH; ++k) {
        int idx = start + k;
        if (idx < n) s += counts[idx];
    }
    part[t] = s;
    __syncthreads();
    for (int off = 1; off < 1024; off <<= 1) {
        int v = part[t];
        int add = (t >= off) ? part[t - off] : 0;
        __syncthreads();
        part[t] = v + add;
        __syncthreads();
    }
    int run = (t == 0) ? 0 : part[t - 1];
    for (int k = 0; k < CH; ++k) {
        int idx = start + k;
        if (idx < n) {
            offsets[idx] = run;
            run += counts[idx];
        }
    }
    if (t == 0) offsets[n] = 0;
}

// ---------------- CSR fill ----------------
__global__ void GNN_fill_kernel(const long long* __restrict__ col,
                                const long long* __restrict__ row,
                                const int* __restrict__ offsets,
                                int* __restrict__ cursor,
                                int* __restrict__ csr, int E) {
    int e = blockIdx.x * blockDim.x + threadIdx.x;
    if (e < E) {
        int c = (int)col[e];
        int p = atomicAdd(&cursor[c], 1);
        csr[offsets[c] + p] = (int)row[e];
    }
}

// ---------------- WMMA gather-mean: one wave32 per node ----------------
// Index blocks are software-pipelined (next block prefetched during current
// block's gather clause); gathers use unsigned 32-bit offsets so the backend
// can emit SGPR-base + voffset addressing. Per 4-edge group:
// D = ones(16x4) x B(4x16) + C per 16-col tile.
__global__ __launch_bounds__(256) void GNN_gather_kernel(
    const float* __restrict__ x, const int* __restrict__ csr,
    const int* __restrict__ offsets, const int* __restrict__ counts,
    float* __restrict__ out, int n_nodes) {
    const int wave = threadIdx.x >> 5;
    const int lane = threadIdx.x & 31;
    const int node = blockIdx.x * 8 + wave;
    if (node >= n_nodes) return;  // wave-uniform

    const int half = lane >> 4;   // B half-wave: K rows {0,1} vs {2,3}
    const int l    = lane & 15;   // N index within a 16-col tile
    const int cnt  = counts[node];
    const int base = offsets[node];

    v2f a; a.x = 1.0f; a.y = 1.0f;  // A = ones(16x4)
    v8f acc[8];
    #pragma unroll
    for (int t = 0; t < 8; ++t)
        acc[t] = (v8f){0.f,0.f,0.f,0.f,0.f,0.f,0.f,0.f};

    if (cnt > 0) {  // wave-uniform; EXEC stays all-ones for WMMA inside
        int idx_cur = csr[base + min(lane, cnt - 1)];
        for (int i0 = 0; i0 < cnt; i0 += 32) {
            const int rem = cnt - i0;  // > 0
            // prefetch next block's indices, clamped & unconditional so the
            // load is in flight during this block's gather clause
            const int posn = min(i0 + 32 + lane, cnt - 1);
            const int idx_next = csr[base + posn];

            if (rem >= 32) {
                #pragma unroll 2
                for (int j = 0; j < 32; j += 4) {
                    const unsigned r0 = (unsigned)__shfl(idx_cur, j + 2 * half, 32);
                    const unsigned r1 = (unsigned)__shfl(idx_cur, j + 2 * half + 1, 32);
                    const unsigned off0 = (r0 << 7) + (unsigned)l;  // element idx
                    const unsigned off1 = (r1 << 7) + (unsigned)l;
                    #pragma unroll
                    for (int t = 0; t < 8; ++t) {
                        v2f b; b.x = x[off0 + t * 16]; b.y = x[off1 + t * 16];
                        acc[t] = __builtin_amdgcn_wmma_f32_16x16x4_f32(
                            false, a, false, b, (short)0, acc[t], false, false);
                    }
                }
            } else {
                // tail: branchless masking via selects (indices pre-clamped)
                for (int j = 0; j < rem; j += 4) {
                    const int k0 = j + 2 * half;
                    const int k1 = k0 + 1;
                    const bool v0 = k0 < rem;
                    const bool v1 = k1 < rem;
                    const unsigned r0 = (unsigned)__shfl(idx_cur, k0, 32);
                    const unsigned r1 = (unsigned)__shfl(idx_cur, k1, 32);
                    const unsigned off0 = (r0 << 7) + (unsigned)l;
                    const unsigned off1 = (r1 << 7) + (unsigned)l;
                    #pragma unroll
                    for (int t = 0; t < 8; ++t) {
                        float bx = x[off0 + t * 16]; if (!v0) bx = 0.0f;
                        float by = x[off1 + t * 16]; if (!v1) by = 0.0f;
                        v2f b; b.x = bx; b.y = by;
                        acc[t] = __builtin_amdgcn_wmma_f32_16x16x4_f32(
                            false, a, false, b, (short)0, acc[t], false, false);
                    }
                }
            }
            idx_cur = idx_next;
        }
    }

    const float scale = 1.0f / (float)(cnt > 0 ? cnt : 1);
    float* orow = out + (size_t)node * (2 * DIM);

    // First half: acc[t][0] (D VGPR0) = colsum for N = lane&15 (rows identical).
    #pragma unroll
    for (int s = 0; s < 4; ++s) {
        float v = half ? acc[s + 4][0] : acc[s][0];
        int c = (half * 4 + s) * 16 + l;
        orow[c] = v * scale;
    }

    // Second half: mean of x[col] over segment == x[node] (or 0 if no edges).
    const float4 xv = ((const float4*)(x + (size_t)node * DIM))[lane];
    float4 res = (cnt > 0) ? xv : make_float4(0.f, 0.f, 0.f, 0.f);
    ((float4*)(orow + DIM))[lane] = res;
}

extern "C" void kernel_launch(void* const* d_in, const int* in_sizes, int n_in,
                              void* d_out, int out_size, void* d_ws, size_t ws_size,
                              hipStream_t stream) {
    (void)n_in; (void)out_size; (void)ws_size;
    const float*     x  = (const float*)d_in[0];
    const long long* es = (const long long*)d_in[1];
    const int N = in_sizes[0] / DIM;   // 10000
    const int E = in_sizes[1] / 2;     // 640000
    const long long* col = es;         // es[0]
    const long long* row = es + E;     // es[1]

    char* ws = (char*)d_ws;
    size_t o = 0;
    auto align256 = [](size_t v) { return (v + 255) & ~(size_t)255; };
    int* counts  = (int*)(ws + o); o += align256((size_t)N * 4);
    int* cursor  = (int*)(ws + o); o += align256((size_t)N * 4);
    int* offsets = (int*)(ws + o); o += align256((size_t)(N + 1) * 4);
    int* csr     = (int*)(ws + o);

    const int TB = 256;
    GNN_zero_kernel<<<(N + TB - 1) / TB, TB, 0, stream>>>(counts, N);
    GNN_zero_kernel<<<(N + TB - 1) / TB, TB, 0, stream>>>(cursor, N);
    GNN_count_kernel<<<(E + TB - 1) / TB, TB, 0, stream>>>(col, counts, E);
    GNN_scan_kernel<<<1, 1024, 0, stream>>>(counts, offsets, N);
    GNN_fill_kernel<<<(E + TB - 1) / TB, TB, 0, stream>>>(col, row, offsets, cursor, csr, E);
    GNN_gather_kernel<<<(N + 7) / 8, TB, 0, stream>>>(x, csr, offsets, counts,
                                                      (float*)d_out, N);
}